// SSDLoss_60060822667518
// MI455X (gfx1250) — compile-verified
//
#include <hip/hip_runtime.h>

// SSD multibox loss for MI455X (gfx1250, wave32).
// One 256-thread workgroup (8 waves) per batch row. Memory-bound:
// ~150 MB total traffic -> ~6.4 us floor at 23.3 TB/s.
// Reductions use V_WMMA_F32_16X16X4_F32 (ones-matmul reduction, exact f32).
// Hard-negative mining done as exact MSB radix-select over LDS keys.

#define A_ANCH 8732
#define NCLS   21
#define BLK    256
#define AITER  ((A_ANCH + BLK - 1) / BLK)   // 35 uniform iterations
#define POSW   ((A_ANCH + 31) / 32)         // 273 words of pos bitmask

typedef __attribute__((ext_vector_type(2))) float v2f;
typedef __attribute__((ext_vector_type(8))) float v8f;

#if __has_builtin(__builtin_amdgcn_wmma_f32_16x16x4_f32)
#define USE_WMMA_F32 1
#endif

__device__ __forceinline__ float sl1(float d) {
    float ad = fabsf(d);
    return ad < 1.0f ? 0.5f * d * d : ad - 0.5f;
}

// Order-preserving float -> uint key (ascending float == ascending uint).
__device__ __forceinline__ unsigned f2key(float f) {
    unsigned u = __float_as_uint(f);
    return (u & 0x80000000u) ? ~u : (u | 0x80000000u);
}

// Block reduction (wave32 shuffles + LDS), uniform control flow only.
__device__ __forceinline__ float blockReduceF(float v, float* s_tmp) {
    __syncthreads();
    for (int o = 16; o; o >>= 1) v += __shfl_xor(v, o, 32);
    const int wid  = threadIdx.x >> 5;
    const int lane = threadIdx.x & 31;
    if (lane == 0) s_tmp[wid] = v;
    __syncthreads();
    float r = 0.0f;
    if (threadIdx.x < (BLK / 32)) r = s_tmp[threadIdx.x];
    if (threadIdx.x < 32) {
        for (int o = 4; o; o >>= 1) r += __shfl_xor(r, o, 32);
    }
    if (threadIdx.x == 0) s_tmp[0] = r;
    __syncthreads();
    return s_tmp[0];
}

__global__ __launch_bounds__(BLK) void ssd_row_kernel(
    const float* __restrict__ lp,   // loc_preds   [N, A, 4]
    const float* __restrict__ lt,   // loc_targets [N, A, 4]
    const float* __restrict__ cp,   // cls_preds   [N, A, C] log-probs
    const int*   __restrict__ ct,   // cls_targets [N, A]
    float* __restrict__ ws)         // ws[0]+=loc+cls, ws[1]+=num_pos
{
    __shared__ float    s_nll[A_ANCH];   // 34928 B
    __shared__ unsigned s_pos[POSW];     //  1092 B
    __shared__ unsigned s_bins[256];     //  1024 B
    __shared__ float    s_tmp[BLK / 32];
    __shared__ unsigned s_prefix, s_krem, s_cless;

    const int n   = blockIdx.x;
    const int tid = threadIdx.x;
    const long long rowBase = (long long)n * A_ANCH;

    for (int i = tid; i < POSW; i += BLK) s_pos[i] = 0u;
    __syncthreads();

    // ---- Phase A: stream everything once ------------------------------
    v8f locC = {0, 0, 0, 0, 0, 0, 0, 0};
    const v2f onesA = {1.0f, 1.0f};
    float locScalar = 0.0f;   // fallback accumulator
    float myNllAll  = 0.0f;
    float myPos     = 0.0f;

    for (int it = 0; it < AITER; ++it) {         // uniform trip count
        const int a = it * BLK + tid;
        const bool valid = (a < A_ANCH);
        float s = 0.0f;
        bool  p = false;
        if (valid) {
            const int t = ct[rowBase + a];
            p = t > 0;
            const float4 d4 = reinterpret_cast<const float4*>(lp)[rowBase + a];
            const float4 t4 = reinterpret_cast<const float4*>(lt)[rowBase + a];
            s = sl1(d4.x - t4.x) + sl1(d4.y - t4.y) +
                sl1(d4.z - t4.z) + sl1(d4.w - t4.w);
            const long long ci = (rowBase + a) * (long long)NCLS;
            __builtin_prefetch(cp + ci + (long long)BLK * NCLS, 0, 0); // global_prefetch_b8
            const float nll = -cp[ci + t];
            s_nll[a] = nll;
            if (p) atomicOr(&s_pos[a >> 5], 1u << (a & 31));
            myNllAll += nll;
            myPos += p ? 1.0f : 0.0f;
        }
        const float lc = (valid && p) ? s : 0.0f;
#ifdef USE_WMMA_F32
        // Ones-matmul reduction: D[i][j] = sum_k B[k][j] + C[i][j]; every
        // column sum appears in d[0] at lanes j and j+16. Uniform EXEC.
        v2f b = {lc, 0.0f};
        locC = __builtin_amdgcn_wmma_f32_16x16x4_f32(
            false, onesA, false, b, (short)0, locC, false, false);
#else
        locScalar += lc;
#endif
    }
    __syncthreads();   // s_nll / s_pos complete

    // ---- Reduce loc, num_pos, total nll -------------------------------
#ifdef USE_WMMA_F32
    float lv = locC[0];
    for (int o = 16; o; o >>= 1) lv += __shfl_xor(lv, o, 32);
    float waveLoc = lv * 0.5f;               // columns duplicated twice
    float locTot = blockReduceF((threadIdx.x & 31) == 0 ? waveLoc : 0.0f, s_tmp);
#else
    float locTot = blockReduceF(locScalar, s_tmp);
#endif
    float nposF  = blockReduceF(myPos, s_tmp);
    float nllAll = blockReduceF(myNllAll, s_tmp);

    const int npos = (int)nposF;
    const long long numneg = 3LL * (long long)npos;

    // ---- Phase B: hard-negative mining --------------------------------
    float clsTot;
    if (numneg >= (long long)A_ANCH) {
        clsTot = nllAll;                      // every anchor selected
    } else if (numneg == 0) {
        clsTot = 0.0f;                        // no positives, no negatives
    } else {
        const unsigned k = (unsigned)numneg;  // take k smallest "mined" keys
        if (tid == 0) { s_prefix = 0u; s_krem = k; s_cless = 0u; }
        __syncthreads();
        for (int pass = 0; pass < 4; ++pass) {
            const int shift = 24 - 8 * pass;
            for (int i = tid; i < 256; i += BLK) s_bins[i] = 0u;
            __syncthreads();
            const unsigned pref  = s_prefix;
            const unsigned pmask = pass == 0 ? 0u : (0xFFFFFFFFu << (32 - 8 * pass));
            for (int a = tid; a < A_ANCH; a += BLK) {
                const bool p = (s_pos[a >> 5] >> (a & 31)) & 1u;
                const unsigned kk = f2key(p ? 0.0f : -s_nll[a]);
                if ((kk & pmask) == pref)
                    atomicAdd(&s_bins[(kk >> shift) & 0xFF], 1u);
            }
            __syncthreads();
            if (tid == 0) {
                unsigned krem = s_krem, cum = 0, dig = 0;
                for (unsigned b2 = 0; b2 < 256; ++b2) {
                    const unsigned c = s_bins[b2];
                    if (cum + c >= krem) { dig = b2; s_krem = krem - cum; s_cless += cum; break; }
                    cum += c;
                }
                s_prefix = pref | (dig << shift);
            }
            __syncthreads();
        }
        const unsigned T = s_prefix;      // k-th smallest key
        const unsigned r = s_krem;        // slots left within the ==T group

        float myCls = 0.0f;
        for (int a = tid; a < A_ANCH; a += BLK) {
            const bool p = (s_pos[a >> 5] >> (a & 31)) & 1u;
            const unsigned kk = f2key(p ? 0.0f : -s_nll[a]);
            if (p || kk < T) myCls += s_nll[a];
        }
        // Stable tie-break among ==T keys: wave 0, ballot prefix-popcount.
        if (tid < 32) {
            unsigned taken = 0; float tieAcc = 0.0f;
            for (int base = 0; base < A_ANCH; base += 32) {
                const int a = base + tid;
                bool eq = false, p = false; float nv = 0.0f;
                if (a < A_ANCH) {
                    p  = (s_pos[a >> 5] >> (a & 31)) & 1u;
                    const unsigned kk = f2key(p ? 0.0f : -s_nll[a]);
                    eq = (kk == T); nv = s_nll[a];
                }
                const unsigned long long m = __ballot(eq);
                const unsigned before = taken +
                    (unsigned)__popcll(m & ((1ull << tid) - 1ull));
                if (eq && before < r && !p) tieAcc += nv;
                taken += (unsigned)__popcll(m);
                if (taken >= r) break;     // uniform across the wave
            }
            myCls += tieAcc;
        }
        clsTot = blockReduceF(myCls, s_tmp);
    }

    if (tid == 0) {
        atomicAdd(ws + 0, locTot + clsTot);
        atomicAdd(ws + 1, nposF);
    }
}

__global__ void ssd_init_ws(float* ws) {
    if (threadIdx.x < 2) ws[threadIdx.x] = 0.0f;
}

__global__ void ssd_finalize(const float* __restrict__ ws, float* __restrict__ out) {
    out[0] = ws[0] / ws[1];
}

extern "C" void kernel_launch(void* const* d_in, const int* in_sizes, int n_in,
                              void* d_out, int out_size, void* d_ws, size_t ws_size,
                              hipStream_t stream) {
    const float* lp = (const float*)d_in[0];
    const float* lt = (const float*)d_in[1];
    const float* cp = (const float*)d_in[2];
    const int*   ct = (const int*)d_in[3];
    float* out = (float*)d_out;
    float* ws  = (float*)d_ws;

    const int N = in_sizes[3] / A_ANCH;   // 128

    ssd_init_ws<<<1, 32, 0, stream>>>(ws);
    ssd_row_kernel<<<N, BLK, 0, stream>>>(lp, lt, cp, ct, ws);
    ssd_finalize<<<1, 1, 0, stream>>>(ws, out);
}